// GraphTransformerLayer_34050500723272
// MI455X (gfx1250) — compile-verified
//
#include <hip/hip_runtime.h>
#include <hip/hip_bf16.h>
#include <math.h>

// ---------------------------------------------------------------------------
// GraphTransformerLayer for MI455X (gfx1250, wave32, WMMA)
//
// Key algebra: softmax is over the HEADS axis, so attn = 1/8 everywhere except
// on edges. attn_out[n] = (1/8)*colsum(v) + sum_{m in adj(n)} (p_h - 1/8) v[m].
// Dense GEMMs use V_WMMA_F32_16X16X4_F32 (full f32 precision, matrix pipe).
// ---------------------------------------------------------------------------

#define NN 4096
#define DD 512
#define DFF 1024
#define NHEAD 8
#define EE 131072
#define BMW 128          // bitmap words per row (4096 bits / 32)

typedef __attribute__((ext_vector_type(2))) float f32x2;
typedef __attribute__((ext_vector_type(8))) float f32x8;

// ------------------------------ zero scratch -------------------------------
__global__ void zero_kernel(unsigned int* __restrict__ bitmap,
                            float* __restrict__ colsum) {
  int i = blockIdx.x * blockDim.x + threadIdx.x;
  if (i < NN * BMW) bitmap[i] = 0u;
  if (i < DD) colsum[i] = 0.0f;
}

// ------------------------------ edge bitmap --------------------------------
__global__ void edge_kernel(const int* __restrict__ ei,
                            unsigned int* __restrict__ bitmap) {
  int e = blockIdx.x * blockDim.x + threadIdx.x;
  if (e < EE) {
    int n = ei[e];        // edge_index[0, e]
    int m = ei[EE + e];   // edge_index[1, e]
    atomicOr(&bitmap[n * BMW + (m >> 5)], 1u << (m & 31));
  }
}

// ------------------------------ column sum of v ----------------------------
__global__ void colsum_kernel(const float* __restrict__ v,
                              float* __restrict__ colsum) {
  int c = threadIdx.x;              // 0..511 (blockDim.x == 512)
  int r0 = blockIdx.x * 256;
  float s = 0.0f;
  for (int m = 0; m < 256; ++m) s += v[(size_t)(r0 + m) * DD + c];
  atomicAdd(&colsum[c], s);
}

// ------------------------------ f32 WMMA GEMM ------------------------------
// C[M,Nc] = A[M,K] @ W[K,Nc] + bias  (optional ReLU).  One wave => 32x32 tile.
// V_WMMA_F32_16X16X4_F32 layouts (ISA 7.12.2):
//   A 16x4 : lane r=lane&15 is row, g=lane>>4; VGPR0=K(2g), VGPR1=K(2g+1)
//   B 4x16 : lane r is col;             VGPR0=K(2g), VGPR1=K(2g+1)
//   C 16x16: VGPR e holds row = e + 8g, col = r
template <int RELU>
__global__ void gemm_kernel(const float* __restrict__ A,
                            const float* __restrict__ W,
                            const float* __restrict__ bias,
                            float* __restrict__ C,
                            int M, int K, int Nc) {
  int wave = (int)((blockIdx.x * blockDim.x + threadIdx.x) >> 5);
  int lane = threadIdx.x & 31;
  int tilesN = Nc >> 5;
  int tM = wave / tilesN;
  int tN = wave % tilesN;
  if (tM * 32 >= M) return;   // wave-uniform; EXEC stays all-ones for WMMA

  int r = lane & 15;
  int g = lane >> 4;

  f32x8 acc00 = {}, acc01 = {}, acc10 = {}, acc11 = {};

  const float* Arow0 = A + (size_t)(tM * 32 + r) * K;
  const float* Arow1 = Arow0 + (size_t)16 * K;
  const float* Wcol = W + tN * 32 + r;

#pragma unroll 4
  for (int k0 = 0; k0 < K; k0 += 4) {
    int ka = k0 + 2 * g;
    f32x2 a0 = { Arow0[ka], Arow0[ka + 1] };
    f32x2 a1 = { Arow1[ka], Arow1[ka + 1] };
    const float* w0 = Wcol + (size_t)ka * Nc;
    const float* w1 = w0 + Nc;
    f32x2 b0 = { w0[0],  w1[0]  };
    f32x2 b1 = { w0[16], w1[16] };
    acc00 = __builtin_amdgcn_wmma_f32_16x16x4_f32(false, a0, false, b0,
                                                  (short)0, acc00, false, false);
    acc01 = __builtin_amdgcn_wmma_f32_16x16x4_f32(false, a0, false, b1,
                                                  (short)0, acc01, false, false);
    acc10 = __builtin_amdgcn_wmma_f32_16x16x4_f32(false, a1, false, b0,
                                                  (short)0, acc10, false, false);
    acc11 = __builtin_amdgcn_wmma_f32_16x16x4_f32(false, a1, false, b1,
                                                  (short)0, acc11, false, false);
  }

  int col0 = tN * 32 + r;
  int col1 = col0 + 16;
  float bias0 = bias[col0];
  float bias1 = bias[col1];
  int row0 = tM * 32 + 8 * g;

#pragma unroll
  for (int e = 0; e < 8; ++e) {
    int ra = row0 + e;
    int rb = ra + 16;
    float v00 = acc00[e] + bias0;
    float v01 = acc01[e] + bias1;
    float v10 = acc10[e] + bias0;
    float v11 = acc11[e] + bias1;
    if (RELU) {
      v00 = fmaxf(v00, 0.0f); v01 = fmaxf(v01, 0.0f);
      v10 = fmaxf(v10, 0.0f); v11 = fmaxf(v11, 0.0f);
    }
    C[(size_t)ra * Nc + col0] = v00;
    C[(size_t)ra * Nc + col1] = v01;
    C[(size_t)rb * Nc + col0] = v10;
    C[(size_t)rb * Nc + col1] = v11;
  }
}

// --------------------- fused sparse attention (head softmax) ---------------
// One wave per node row n. Each lane owns 16 contiguous channels
// (lane covers head h = lane/4). acc starts at (1/8)*colsum(v); every
// neighbor m contributes (p_h - 1/8) * v[m].
__global__ void attn_kernel(const float* __restrict__ q,
                            const float* __restrict__ k,
                            const float* __restrict__ v,
                            const unsigned int* __restrict__ bitmap,
                            const float* __restrict__ colsum,
                            float* __restrict__ out) {
  int n = (int)((blockIdx.x * blockDim.x + threadIdx.x) >> 5);
  int lane = threadIdx.x & 31;
  int c0 = lane * 16;

  float qr[16];
  float acc[16];
  const float* qp = q + (size_t)n * DD + c0;
#pragma unroll
  for (int j = 0; j < 16; ++j) {
    qr[j] = qp[j];
    acc[j] = 0.125f * colsum[c0 + j];
  }

  const unsigned int* row = bitmap + n * BMW;
  for (int w = 0; w < BMW; ++w) {
    unsigned int bits = row[w];
    while (bits) {
      int b = __ffs(bits) - 1;
      bits &= bits - 1;
      int m = (w << 5) + b;
      const float* kp = k + (size_t)m * DD + c0;
      const float* vp = v + (size_t)m * DD + c0;

      float dotp = 0.0f;
#pragma unroll
      for (int j = 0; j < 16; ++j) dotp += qr[j] * kp[j];
      // reduce the 4 lanes of this head (lane bits 0..1)
      dotp += __shfl_xor(dotp, 1, 32);
      dotp += __shfl_xor(dotp, 2, 32);
      float s = dotp * 0.125f;                 // 1/sqrt(64)

      // softmax across the 8 heads (lane bits 2..4)
      float mx = s;
      mx = fmaxf(mx, __shfl_xor(mx, 4, 32));
      mx = fmaxf(mx, __shfl_xor(mx, 8, 32));
      mx = fmaxf(mx, __shfl_xor(mx, 16, 32));
      float ex = __expf(s - mx);
      float sm = ex;
      sm += __shfl_xor(sm, 4, 32);
      sm += __shfl_xor(sm, 8, 32);
      sm += __shfl_xor(sm, 16, 32);
      float wgt = ex / sm - 0.125f;            // replace the uniform 1/8

#pragma unroll
      for (int j = 0; j < 16; ++j) acc[j] += wgt * vp[j];
    }
  }

  float* op = out + (size_t)n * DD + c0;
#pragma unroll
  for (int j = 0; j < 16; ++j) op[j] = acc[j];
}

// ------------------------------ LayerNorm ----------------------------------
// out = LN(a + b) * gamma + beta, one wave per row of 512.
__global__ void ln_kernel(const float* __restrict__ a,
                          const float* __restrict__ b,
                          const float* __restrict__ gamma,
                          const float* __restrict__ beta,
                          float* __restrict__ out) {
  int n = (int)((blockIdx.x * blockDim.x + threadIdx.x) >> 5);
  int lane = threadIdx.x & 31;
  int c0 = lane * 16;
  const float* ap = a + (size_t)n * DD + c0;
  const float* bp = b + (size_t)n * DD + c0;

  float xv[16];
  float s = 0.0f;
#pragma unroll
  for (int j = 0; j < 16; ++j) {
    xv[j] = ap[j] + bp[j];
    s += xv[j];
  }
  s += __shfl_xor(s, 1, 32);
  s += __shfl_xor(s, 2, 32);
  s += __shfl_xor(s, 4, 32);
  s += __shfl_xor(s, 8, 32);
  s += __shfl_xor(s, 16, 32);
  float mean = s * (1.0f / DD);

  float var = 0.0f;
#pragma unroll
  for (int j = 0; j < 16; ++j) {
    float d = xv[j] - mean;
    var += d * d;
  }
  var += __shfl_xor(var, 1, 32);
  var += __shfl_xor(var, 2, 32);
  var += __shfl_xor(var, 4, 32);
  var += __shfl_xor(var, 8, 32);
  var += __shfl_xor(var, 16, 32);
  float inv = rsqrtf(var * (1.0f / DD) + 1e-5f);

  float* op = out + (size_t)n * DD + c0;
#pragma unroll
  for (int j = 0; j < 16; ++j)
    op[j] = (xv[j] - mean) * inv * gamma[c0 + j] + beta[c0 + j];
}

// ------------------------------ launch -------------------------------------
extern "C" void kernel_launch(void* const* d_in, const int* in_sizes, int n_in,
                              void* d_out, int out_size, void* d_ws, size_t ws_size,
                              hipStream_t stream) {
  const float* x  = (const float*)d_in[0];
  const int*   ei = (const int*)  d_in[1];
  const float* Wq = (const float*)d_in[2];
  const float* bq = (const float*)d_in[3];
  const float* Wk = (const float*)d_in[4];
  const float* bk = (const float*)d_in[5];
  const float* Wv = (const float*)d_in[6];
  const float* bv = (const float*)d_in[7];
  const float* Wo = (const float*)d_in[8];
  const float* bo = (const float*)d_in[9];
  const float* W1 = (const float*)d_in[10];
  const float* b1 = (const float*)d_in[11];
  const float* W2 = (const float*)d_in[12];
  const float* b2 = (const float*)d_in[13];
  const float* g1 = (const float*)d_in[14];
  const float* be1= (const float*)d_in[15];
  const float* g2 = (const float*)d_in[16];
  const float* be2= (const float*)d_in[17];
  float* out = (float*)d_out;

  float* ws = (float*)d_ws;
  float* q    = ws;                         // NN*DD
  float* kk   = q    + (size_t)NN * DD;     // NN*DD
  float* v    = kk   + (size_t)NN * DD;     // NN*DD
  float* attn = v    + (size_t)NN * DD;     // NN*DD
  float* ao   = attn + (size_t)NN * DD;     // NN*DD
  float* h1   = ao   + (size_t)NN * DD;     // NN*DD
  float* ff1  = h1   + (size_t)NN * DD;     // NN*DFF
  float* ff2  = ff1  + (size_t)NN * DFF;    // NN*DD
  float* colsum = ff2 + (size_t)NN * DD;    // DD
  unsigned int* bitmap = (unsigned int*)(colsum + DD);  // NN*BMW words

  // 1) zero bitmap + colsum (ws is not re-poisoned between replays)
  {
    int total = NN * BMW;
    zero_kernel<<<(total + 255) / 256, 256, 0, stream>>>(bitmap, colsum);
  }

  // 2) q/k/v projections (f32 WMMA). waves = (4096/32)*(512/32) = 2048
  gemm_kernel<0><<<256, 256, 0, stream>>>(x, Wq, bq, q,  NN, DD, DD);
  gemm_kernel<0><<<256, 256, 0, stream>>>(x, Wk, bk, kk, NN, DD, DD);
  gemm_kernel<0><<<256, 256, 0, stream>>>(x, Wv, bv, v,  NN, DD, DD);

  // 3) adjacency bitmap (dedupes duplicate edges)
  edge_kernel<<<EE / 256, 256, 0, stream>>>(ei, bitmap);

  // 4) column sum of v for the uniform-1/8 background term
  colsum_kernel<<<NN / 256, DD, 0, stream>>>(v, colsum);

  // 5) fused sparse attention with head-axis softmax. 4096 waves.
  attn_kernel<<<NN / 8, 256, 0, stream>>>(q, kk, v, bitmap, colsum, attn);

  // 6) output projection
  gemm_kernel<0><<<256, 256, 0, stream>>>(attn, Wo, bo, ao, NN, DD, DD);

  // 7) h1 = LN(x + ao)
  ln_kernel<<<NN / 8, 256, 0, stream>>>(x, ao, g1, be1, h1);

  // 8) FFN: relu(h1@W1+b1) @ W2 + b2. waves = 128*32 = 4096, then 128*16 = 2048
  gemm_kernel<1><<<512, 256, 0, stream>>>(h1,  W1, b1, ff1, NN, DD,  DFF);
  gemm_kernel<0><<<256, 256, 0, stream>>>(ff1, W2, b2, ff2, NN, DFF, DD);

  // 9) out = LN(h1 + ff2)
  ln_kernel<<<NN / 8, 256, 0, stream>>>(h1, ff2, g2, be2, out);
}